// QEmbedding_26027501814040
// MI455X (gfx1250) — compile-verified
//
#include <hip/hip_runtime.h>
#include <hip/hip_bf16.h>

// ---------------------------------------------------------------------------
// QEmbedding on MI455X (gfx1250, wave32)
//
// psi = L2 * CNOTring * L1 * |0>, table = |psi| (8192x256), out = LN(table[x])
//
// K0: setup   - partial-product tables for the fused (L1+CNOT) state,
//               16x16 group unitaries for L2 (wires 0-3,...,16-19), U20.
// K1: build   - psi[y] = T0*T1*T2 at bits g^{-1}(y)   (CNOT ring inverted
//               in closed form: x = (y^(y>>1)) on low bits + 2 fixups)
// K2 x5: WMMA - 16x16 complex matmul per 4-wire group via
//               V_WMMA_F32_16X16X4_F32 chains (K=16 -> 4 chained wmma,
//               4 real products RR/II/RI/IR per complex GEMM).
// K3: gather  - fuse wire-20 2x2 gate + |.| + two-pass LayerNorm(256).
// ---------------------------------------------------------------------------

typedef float v2f __attribute__((ext_vector_type(2)));
typedef float v8f __attribute__((ext_vector_type(8)));

#define NW 21
#define NSTATE (1u << NW)   // 2^21 amplitudes

__device__ __forceinline__ float2 cmul(float2 a, float2 b) {
    return make_float2(a.x * b.x - a.y * b.y, a.x * b.y + a.y * b.x);
}

// Rot = RZ(omega) RY(theta) RZ(phi):
//   m00 = e^{-i(phi+omega)/2} c ; m01 = -e^{ i(phi-omega)/2} s
//   m10 = e^{-i(phi-omega)/2} s ; m11 =  e^{ i(phi+omega)/2} c
__device__ __forceinline__ void rot_mat(const float* rp, int i, float2 m[2][2]) {
    float phi = rp[3 * i + 0], th = rp[3 * i + 1], om = rp[3 * i + 2];
    float ch = cosf(th * 0.5f), sh = sinf(th * 0.5f);
    float a = 0.5f * (phi + om), b = 0.5f * (phi - om);
    float sa, ca, sb, cb;
    sincosf(a, &sa, &ca);
    sincosf(b, &sb, &cb);
    m[0][0] = make_float2( ca * ch, -sa * ch);
    m[0][1] = make_float2(-cb * sh, -sb * sh);
    m[1][0] = make_float2( cb * sh, -sb * sh);
    m[1][1] = make_float2( ca * ch,  sa * ch);
}

// K0 -------------------------------------------------------------------------
// T  : 3 tables x 128 entries (float2 interleaved) -> 768 floats
//      T_t[idx] = prod_{d=0..6} U1_{7t+d}[ bit_(6-d)(idx) ][0]
// G  : 5 groups x (Gr[256] ++ Gi[256]) row-major 16x16 -> 2560 floats
//      G_g[j][k] = prod_{d=0..3} U2_{4g+d}[ bit_(3-d)(j) ][ bit_(3-d)(k) ]
// U20: 8 floats  (re00 im00 re01 im01 re10 im10 re11 im11)
__global__ void qsetup_kernel(const float* __restrict__ rp1,
                              const float* __restrict__ rp2,
                              float* __restrict__ T,
                              float* __restrict__ G,
                              float* __restrict__ u20) {
    int tid = threadIdx.x;
    for (int e = tid; e < 384; e += 256) {
        int t = e >> 7, idx = e & 127;
        float2 acc = make_float2(1.0f, 0.0f);
        for (int d = 0; d < 7; ++d) {
            int bit = (idx >> (6 - d)) & 1;
            float2 m[2][2];
            rot_mat(rp1, 7 * t + d, m);
            acc = cmul(acc, m[bit][0]);
        }
        T[2 * e + 0] = acc.x;
        T[2 * e + 1] = acc.y;
    }
    for (int e = tid; e < 1280; e += 256) {
        int g = e >> 8, jk = e & 255, j = jk >> 4, k = jk & 15;
        float2 acc = make_float2(1.0f, 0.0f);
        for (int d = 0; d < 4; ++d) {
            float2 m[2][2];
            rot_mat(rp2, 4 * g + d, m);
            acc = cmul(acc, m[(j >> (3 - d)) & 1][(k >> (3 - d)) & 1]);
        }
        G[g * 512 + j * 16 + k]       = acc.x;  // Gr
        G[g * 512 + 256 + j * 16 + k] = acc.y;  // Gi
    }
    if (tid == 0) {
        float2 m[2][2];
        rot_mat(rp2, 20, m);
        u20[0] = m[0][0].x; u20[1] = m[0][0].y;
        u20[2] = m[0][1].x; u20[3] = m[0][1].y;
        u20[4] = m[1][0].x; u20[5] = m[1][0].y;
        u20[6] = m[1][1].x; u20[7] = m[1][1].y;
    }
}

// K1 -------------------------------------------------------------------------
// psi2[y] = psi1[h(y)] where h inverts the CNOT ring (wire i -> bit 20-i):
//   low 19 bits: x = (y ^ (y>>1)); bit19: y19^y20^y0pos; bit20: y20^y0pos
__global__ void qbuild_state(const float* __restrict__ T,
                             float* __restrict__ re,
                             float* __restrict__ im) {
    __shared__ float2 lt[384];
    const float2* tp = (const float2*)T;
    int tid = threadIdx.x;
    for (int e = tid; e < 384; e += 256) lt[e] = tp[e];
    __syncthreads();

    unsigned y = blockIdx.x * 256u + (unsigned)tid;
    unsigned lo  = (y ^ (y >> 1)) & 0x7FFFFu;
    unsigned b19 = ((y >> 19) ^ (y >> 20) ^ y) & 1u;
    unsigned b20 = ((y >> 20) ^ y) & 1u;
    unsigned x = lo | (b19 << 19) | (b20 << 20);

    float2 a = cmul(lt[(x >> 14) & 127u], lt[128 + ((x >> 7) & 127u)]);
    a = cmul(a, lt[256 + (x & 127u)]);
    re[y] = a.x;
    im[y] = a.y;
}

// K2 -------------------------------------------------------------------------
// One 4-wire group (wires w..w+3): psi'[o,j,i] = sum_k G[j,k] psi[o,k,i]
// Flattened columns c in [0, 2^17): o = c >> log2I, i = c & (I-1),
// element addr = (o << (log2I+4)) + (k << log2I) + i.  One wave per 16-col
// tile; complex GEMM = 4 real chained-WMMA products.
__global__ void qgate4_wmma(float* __restrict__ re, float* __restrict__ im,
                            const float* __restrict__ Gr,
                            const float* __restrict__ Gi, int w) {
    const int lane = threadIdx.x & 31;
    const int wave = (int)((blockIdx.x * blockDim.x + threadIdx.x) >> 5);  // 0..8191
    const int log2I = 17 - w;
    const unsigned Imask = (1u << log2I) - 1u;
    const int m    = lane & 15;   // A row / B,D column-in-tile
    const int half = lane >> 4;
    const unsigned c = ((unsigned)wave << 4) + (unsigned)m;
    const unsigned o = c >> log2I;
    const unsigned i = c & Imask;
    const unsigned colAddr = (o << (log2I + 4)) + i;

    // A operands: lane L, vgpr v -> A[m][k], k = 4q + 2*half + v
    v2f ar[4], ai[4];
#pragma unroll
    for (int q = 0; q < 4; ++q) {
        int k0 = 4 * q + 2 * half;
        ar[q][0] = Gr[m * 16 + k0];     ar[q][1] = Gr[m * 16 + k0 + 1];
        ai[q][0] = Gi[m * 16 + k0];     ai[q][1] = Gi[m * 16 + k0 + 1];
    }
    // B operands: lane L, vgpr v -> B[k][c], same k mapping, col = c
    v2f br[4], bi[4];
#pragma unroll
    for (int q = 0; q < 4; ++q) {
        unsigned k0 = (unsigned)(4 * q + 2 * half);
        br[q][0] = re[colAddr + (k0 << log2I)];
        br[q][1] = re[colAddr + ((k0 + 1u) << log2I)];
        bi[q][0] = im[colAddr + (k0 << log2I)];
        bi[q][1] = im[colAddr + ((k0 + 1u) << log2I)];
    }

    v8f rr = {}, ii = {}, ri = {}, ir = {};
#pragma unroll
    for (int q = 0; q < 4; ++q) {
        rr = __builtin_amdgcn_wmma_f32_16x16x4_f32(false, ar[q], false, br[q],
                                                   (short)0, rr, false, false);
        ii = __builtin_amdgcn_wmma_f32_16x16x4_f32(false, ai[q], false, bi[q],
                                                   (short)0, ii, false, false);
        ri = __builtin_amdgcn_wmma_f32_16x16x4_f32(false, ar[q], false, bi[q],
                                                   (short)0, ri, false, false);
        ir = __builtin_amdgcn_wmma_f32_16x16x4_f32(false, ai[q], false, br[q],
                                                   (short)0, ir, false, false);
    }
    v8f dre = rr - ii;   // Re' = Gr*Re - Gi*Im
    v8f dim = ri + ir;   // Im' = Gr*Im + Gi*Re

    // D: vgpr r -> row j = r + 8*half, col = c
#pragma unroll
    for (int r = 0; r < 8; ++r) {
        unsigned j = (unsigned)(r + 8 * half);
        unsigned addr = colAddr + (j << log2I);
        re[addr] = dre[r];
        im[addr] = dim[r];
    }
}

// K3 -------------------------------------------------------------------------
// One block (256 thr) per token: fuse wire-20 gate + abs + LayerNorm.
__global__ void qgather_ln(const int* __restrict__ xi,
                           const float* __restrict__ re,
                           const float* __restrict__ im,
                           const float* __restrict__ u20,
                           const float* __restrict__ gamma,
                           const float* __restrict__ beta,
                           float* __restrict__ out) {
    __shared__ float red[8];
    __shared__ float sb;
    const int token = blockIdx.x;
    const int e = threadIdx.x;
    const unsigned base = ((unsigned)xi[token]) << 8;

    const int p = e >> 1, t = e & 1;
    float er0 = re[base + 2 * p],     ei0 = im[base + 2 * p];
    float er1 = re[base + 2 * p + 1], ei1 = im[base + 2 * p + 1];
    float ur0 = u20[t * 4 + 0], ui0 = u20[t * 4 + 1];
    float ur1 = u20[t * 4 + 2], ui1 = u20[t * 4 + 3];
    float fr = ur0 * er0 - ui0 * ei0 + ur1 * er1 - ui1 * ei1;
    float fi = ur0 * ei0 + ui0 * er0 + ur1 * ei1 + ui1 * er1;
    float v = sqrtf(fr * fr + fi * fi);

    // mean
    float s = v;
    for (int off = 16; off; off >>= 1) s += __shfl_xor(s, off, 32);
    if ((e & 31) == 0) red[e >> 5] = s;
    __syncthreads();
    if (e == 0) {
        float acc = 0.f;
        for (int k = 0; k < 8; ++k) acc += red[k];
        sb = acc * (1.0f / 256.0f);
    }
    __syncthreads();
    float mu = sb;
    float d = v - mu;

    // variance (two-pass, matches reference form)
    float s2 = d * d;
    for (int off = 16; off; off >>= 1) s2 += __shfl_xor(s2, off, 32);
    if ((e & 31) == 0) red[e >> 5] = s2;
    __syncthreads();
    if (e == 0) {
        float acc = 0.f;
        for (int k = 0; k < 8; ++k) acc += red[k];
        sb = acc * (1.0f / 256.0f);
    }
    __syncthreads();
    float rs = rsqrtf(sb + 1e-5f);

    out[(unsigned)token * 256u + (unsigned)e] = d * rs * gamma[e] + beta[e];
}

// ---------------------------------------------------------------------------
extern "C" void kernel_launch(void* const* d_in, const int* in_sizes, int n_in,
                              void* d_out, int out_size, void* d_ws, size_t ws_size,
                              hipStream_t stream) {
    const int*   x     = (const int*)d_in[0];
    const float* rp1   = (const float*)d_in[1];
    const float* rp2   = (const float*)d_in[2];
    const float* gamma = (const float*)d_in[3];
    const float* beta  = (const float*)d_in[4];
    float*       out   = (float*)d_out;

    float* ws  = (float*)d_ws;
    float* re  = ws;                       // 2^21 floats (8 MB)
    float* im  = ws + NSTATE;              // 2^21 floats (8 MB)
    float* T   = ws + 2 * NSTATE;          // 768 floats
    float* G   = T + 768;                  // 2560 floats
    float* u20 = G + 2560;                 // 8 floats

    qsetup_kernel<<<1, 256, 0, stream>>>(rp1, rp2, T, G, u20);
    qbuild_state<<<NSTATE / 256, 256, 0, stream>>>(T, re, im);
    for (int g = 0; g < 5; ++g) {
        qgate4_wmma<<<1024, 256, 0, stream>>>(re, im, G + g * 512,
                                              G + g * 512 + 256, 4 * g);
    }
    qgather_ln<<<16384, 256, 0, stream>>>(x, re, im, u20, gamma, beta, out);
}